// ApproachingMomentumLoss_12446815224159
// MI455X (gfx1250) — compile-verified
//
#include <hip/hip_runtime.h>

typedef __attribute__((ext_vector_type(2))) float v2f;
typedef __attribute__((ext_vector_type(8))) float v8f;

#define TLEN 4096
#define RADIUS_I 20

// Uniform-lane broadcast via v_readlane_b32 (constant lane, all lanes active)
// instead of HIP __shfl's ds_bpermute path — removes LDS round-trips from the
// WMMA -> row-totals -> WMMA critical path.
static __device__ __forceinline__ float readlane_f(float v, int srclane) {
    return __int_as_float(__builtin_amdgcn_readlane(__float_as_int(v), srclane));
}

// One wave32 per row. WMMA-based matmul-scan for the cumsum, LDS window scan
// for the clamped distance transform, wave-shuffle reductions for the loss.
__global__ __launch_bounds__(32) void aml_row_kernel(
    const float* __restrict__ vel,
    const unsigned char* __restrict__ bnd,   // jax bool: 1 byte/elem
    const unsigned char* __restrict__ msk,   // jax bool: 1 byte/elem
    float* __restrict__ ws)
{
    __shared__ unsigned char b_lds[TLEN];    // boundaries | ~mask
    __shared__ unsigned char m_lds[TLEN];    // mask

    const int row  = blockIdx.x;
    const int lane = threadIdx.x;
    const float*         vrowp = vel + (size_t)row * TLEN;
    const unsigned char* brow  = bnd + (size_t)row * TLEN;
    const unsigned char* mrow  = msk + (size_t)row * TLEN;

    // Stage boundary/mask bytes into LDS (effective boundary = b | !mask).
    for (int i = lane; i < TLEN; i += 32) {
        unsigned char m = mrow[i];
        unsigned char b = brow[i];
        b_lds[i] = (unsigned char)((b != 0) || (m == 0));
        m_lds[i] = (unsigned char)(m != 0);
    }
    __syncthreads();

    const int n = lane & 15;           // N column (B/D) and M row (A) index
    const bool lo = (lane < 16);       // lane-group select for K split

    float carry = 0.0f;                // running cumsum across 256-chunks
    float lsum  = 0.0f;                // sum |pred - gt| * mask (this lane)
    float gmax  = 0.0f;                // max gt (this lane)
    int   mcount = 0;                  // mask count (this lane)

    for (int chunk = 0; chunk < 16; ++chunk) {
        const int base = chunk * 256;

        // Lane (and lane+16) holds row m=n of V: 16 masked velocities.
        float vr[16];
        const int off = base + 16 * n;
#pragma unroll
        for (int k = 0; k < 16; ++k)
            vr[k] = vrowp[off + k] * (float)m_lds[off + k];

        // Stage 1: P = V x U (U upper-triangular incl. diag), K=16 as 4x K=4.
        v8f acc = {};
#pragma unroll
        for (int kc = 0; kc < 4; ++kc) {
            const int k0 = 4 * kc + (lo ? 0 : 2);
            const int k1 = k0 + 1;
            v2f a, b;
            a.x = lo ? vr[4 * kc + 0] : vr[4 * kc + 2];
            a.y = lo ? vr[4 * kc + 1] : vr[4 * kc + 3];
            b.x = (k0 <= n) ? 1.0f : 0.0f;
            b.y = (k1 <= n) ? 1.0f : 0.0f;
            acc = __builtin_amdgcn_wmma_f32_16x16x4_f32(
                false, a, false, b, (short)0, acc, false, false);
        }

        // Row totals t[k] = P[k][15]  (C/D layout: lane 15 / lane 31),
        // broadcast with v_readlane_b32 (uniform constant source lane).
        float t[16];
#pragma unroll
        for (int k = 0; k < 16; ++k)
            t[k] = readlane_f(acc[k & 7], (k < 8) ? 15 : 31);

        // Stage 2: X = Lstrict x Trep + P  (cross-row offsets).
        v8f x = acc;
#pragma unroll
        for (int kc = 0; kc < 4; ++kc) {
            const int k0 = 4 * kc + (lo ? 0 : 2);
            const int k1 = k0 + 1;
            v2f a, b;
            a.x = (k0 < n) ? 1.0f : 0.0f;
            a.y = (k1 < n) ? 1.0f : 0.0f;
            b.x = lo ? t[4 * kc + 0] : t[4 * kc + 2];
            b.y = lo ? t[4 * kc + 1] : t[4 * kc + 3];
            x = __builtin_amdgcn_wmma_f32_16x16x4_f32(
                false, a, false, b, (short)0, x, false, false);
        }

        const float chunk_total = readlane_f(x[7], 31);  // X[15][15]

        // Loss accumulation: element (g,lane) -> pos = base + 16*M + N.
#pragma unroll
        for (int g = 0; g < 8; ++g) {
            const int m   = g + (lo ? 0 : 8);
            const int pos = base + 16 * m + n;
            const float pred = x[g] + carry;

            // gt distance: min(20, dist to nearest effective boundary,
            // sentinels at -1 and TLEN).
            int dmin = RADIUS_I;
            dmin = min(dmin, pos + 1);
            dmin = min(dmin, TLEN - pos);
            for (int d = 0; d < dmin; ++d) {
                if (b_lds[pos - d] | b_lds[pos + d]) { dmin = d; break; }
            }
            const float gt = (float)dmin;
            gmax = fmaxf(gmax, gt);
            const float mf = (float)m_lds[pos];
            lsum += fabsf(pred - gt) * mf;
            mcount += (int)m_lds[pos];
        }
        carry += chunk_total;
    }

    // Wave32 reductions.
#pragma unroll
    for (int s = 16; s > 0; s >>= 1) {
        lsum   += __shfl_xor(lsum, s, 32);
        gmax    = fmaxf(gmax, __shfl_xor(gmax, s, 32));
        mcount += __shfl_xor(mcount, s, 32);
    }
    if (lane == 0) {
        ws[row * 2 + 0] = lsum / (gmax + 1e-6f);   // row loss (already /scale)
        ws[row * 2 + 1] = (float)mcount;
    }
}

__global__ __launch_bounds__(32) void aml_finalize_kernel(
    const float* __restrict__ ws, float* __restrict__ out, int nrows)
{
    if (threadIdx.x == 0) {
        float s = 0.0f, c = 0.0f;
        for (int r = 0; r < nrows; ++r) {
            s += ws[2 * r + 0];
            c += ws[2 * r + 1];
        }
        out[0] = s / (c + 1e-6f);
    }
}

extern "C" void kernel_launch(void* const* d_in, const int* in_sizes, int n_in,
                              void* d_out, int out_size, void* d_ws, size_t ws_size,
                              hipStream_t stream) {
    const float*         vel = (const float*)d_in[0];
    const unsigned char* bnd = (const unsigned char*)d_in[1];
    const unsigned char* msk = (const unsigned char*)d_in[2];
    float* ws  = (float*)d_ws;
    float* out = (float*)d_out;

    const int nrows = in_sizes[0] / TLEN;   // B = 8

    aml_row_kernel<<<nrows, 32, 0, stream>>>(vel, bnd, msk, ws);
    aml_finalize_kernel<<<1, 32, 0, stream>>>(ws, out, nrows);
}